// SAGE_84146999263864
// MI455X (gfx1250) — compile-verified
//
#include <hip/hip_runtime.h>

// ---------------- problem constants (match reference) ----------------
#define N_NODES 50000
#define N_EDGES 800000
#define N_FEAT  128
#define N_HID   128
#define N_CLS   40
#define N_CLS_P 48          // padded to 3 WMMA col-tiles
#define BN_EPS  1e-5f

typedef __attribute__((ext_vector_type(2))) float v2f;
typedef __attribute__((ext_vector_type(8))) float v8f;
typedef __attribute__((ext_vector_type(4))) unsigned int v4u;
typedef __attribute__((ext_vector_type(4))) int v4i;
typedef __attribute__((ext_vector_type(8))) int v8i;

// ---------------- workspace layout (floats) ----------------
// [invdeg: 50048][agg1: 6.4M][agg2: 6.4M][h: 6.4M][W2l_p: 6144][W2r_p: 6144][b2_p: 48]
#define WS_INVDEG 0
#define WS_AGG1   50048
#define WS_AGG2   (WS_AGG1 + N_NODES * N_FEAT)
#define WS_H      (WS_AGG2 + N_NODES * N_HID)
#define WS_W2L    (WS_H    + N_NODES * N_HID)
#define WS_W2R    (WS_W2L  + N_CLS_P * N_HID)
#define WS_B2     (WS_W2R  + N_CLS_P * N_HID)
#define WS_ZERO_F (WS_AGG2 + N_NODES * N_HID)   // invdeg+agg1+agg2 contiguous

// ---------------- TDM: async-load a 16x128 f32 tile (row stride 128) into LDS ----------
// D# per CDNA5 ISA ch.8: group0 = {count, lds_addr, global_addr[56:0], type=2},
// group1 = {data_size=4B, tensor_dim0=128, tensor_dim1=16, tile_dim0=128, tile_dim1=16,
//           tensor_dim0_stride=128}. Groups 2/3 zero (tensor <= 2D).
__device__ __forceinline__ void tdm_load_tile_16x128(const float* gptr, unsigned lds_byte_off) {
    unsigned long long ga = (unsigned long long)gptr;
    v4u g0;
    g0.x = 1u;                                            // count=1 (valid user D#)
    g0.y = lds_byte_off;                                  // LDS byte address of tile
    g0.z = (unsigned)(ga & 0xffffffffull);                // global_addr[31:0]
    g0.w = (unsigned)((ga >> 32) & 0x1ffffffull)          // global_addr[56:32]
         | (2u << 30);                                    // type=2 ("image")
    v8i g1;
    g1[0] = (int)(2u << 16);                              // data_size=2 -> 4 bytes
    g1[1] = (int)((128u & 0xffffu) << 16);                // tensor_dim0[15:0]=128
    g1[2] = (int)((16u & 0xffffu) << 16);                 // tensor_dim0[31:16]=0 | tensor_dim1[15:0]=16
    g1[3] = (int)(128u << 16);                            // tensor_dim1[31:16]=0 | tile_dim0=128
    g1[4] = (int)16u;                                     // tile_dim1=16, tile_dim2=0
    g1[5] = (int)128u;                                    // tensor_dim0_stride[31:0]=128 elements
    g1[6] = 0;                                            // stride0[47:32] | stride1[15:0]
    g1[7] = 0;                                            // stride1[47:16]
    v4i z4 = {0, 0, 0, 0};
    v8i z8 = {0, 0, 0, 0, 0, 0, 0, 0};
    // 6-arg toolchain form: (g0, g1, g2, g3, g4, cpol)
    __builtin_amdgcn_tensor_load_to_lds(g0, g1, z4, z4, z8, 0);
}

// ---------------- 1) zero accumulators (float4) ----------------
__global__ void sage_zero_kernel(float* __restrict__ p, int n4) {
    int i = blockIdx.x * blockDim.x + threadIdx.x;
    if (i < n4) ((float4*)p)[i] = make_float4(0.f, 0.f, 0.f, 0.f);
}

// ---------------- 2) degree via f32 atomics ----------------
__global__ void sage_deg_kernel(const int* __restrict__ dst, float* __restrict__ deg) {
    int e = blockIdx.x * blockDim.x + threadIdx.x;
    if (e < N_EDGES) atomicAdd(&deg[dst[e]], 1.0f);
}

__global__ void sage_invdeg_kernel(float* __restrict__ deg) {
    int i = blockIdx.x * blockDim.x + threadIdx.x;
    if (i < N_NODES) deg[i] = 1.0f / fmaxf(deg[i], 1.0f);
}

// ---------------- 3) edge scatter: agg[dst] += feat[src] ----------------
// one thread per (edge, 4-float feature group): 800000*32 threads
__global__ __launch_bounds__(256) void sage_scatter_kernel(
    const int* __restrict__ src, const int* __restrict__ dst,
    const float* __restrict__ feat, float* __restrict__ agg) {
    int t = blockIdx.x * blockDim.x + threadIdx.x;
    int e = t >> 5;
    if (e >= N_EDGES) return;
    int g = (t & 31) << 2;
    int s = src[e], d = dst[e];
    float4 v = *(const float4*)(feat + s * N_FEAT + g);
    float* o = agg + d * N_FEAT + g;
    atomicAdd(o + 0, v.x);
    atomicAdd(o + 1, v.y);
    atomicAdd(o + 2, v.z);
    atomicAdd(o + 3, v.w);
}

// ---------------- 4) fused GEMM1: h = BN(relu(agg*invdeg @ W1l^T + b1 + x @ W1r^T)) ----------------
// Block = 8 waves sharing one 16-row tile; TDM stages the agg & x 16x128 tiles into LDS,
// each wave owns one 16-col weight tile and reads A fragments from LDS.
__global__ __launch_bounds__(256) void sage_gemm1_kernel(
    const float* __restrict__ agg, const float* __restrict__ x,
    const float* __restrict__ W1l, const float* __restrict__ W1r,
    const float* __restrict__ b1,  const float* __restrict__ gamma,
    const float* __restrict__ beta, const float* __restrict__ mean,
    const float* __restrict__ var,  const float* __restrict__ invdeg,
    float* __restrict__ h) {
    extern __shared__ float smem[];           // [0,2048): agg tile, [2048,4096): x tile
    const int lane = threadIdx.x & 31;
    const int wave = threadIdx.x >> 5;        // 0..7 -> col tile
    const int m0 = blockIdx.x * 16;           // row tile (3125 blocks, exact)
    const int n0 = wave * 16;                 // 8 col tiles cover N_HID=128
    const int lrow  = lane & 15;
    const int khalf = (lane >> 4) << 1;       // 0 or 2: K-pair this lane holds

    // stage A tiles via Tensor Data Mover (one descriptor per tile)
    if (wave == 0) {
        tdm_load_tile_16x128(agg + m0 * N_FEAT, 0u);
        __builtin_amdgcn_s_wait_tensorcnt(0);
    } else if (wave == 1) {
        tdm_load_tile_16x128(x + m0 * N_FEAT, 16u * N_FEAT * 4u);
        __builtin_amdgcn_s_wait_tensorcnt(0);
    }
    __syncthreads();

    const int brow = n0 + lrow;               // weight row (output feature)
    const float idg = invdeg[m0 + lrow];

    const float* aggl = smem + lrow * N_FEAT + khalf;             // LDS
    const float* xl   = smem + 16 * N_FEAT + lrow * N_FEAT + khalf;
    const float* wlp  = W1l + brow * N_FEAT + khalf;              // global
    const float* wrp  = W1r + brow * N_FEAT + khalf;

    v8f c = {};
#pragma unroll
    for (int k = 0; k < N_FEAT; k += 4) {
        v2f a0 = *(const v2f*)(aggl + k);
        a0 *= idg;                            // fold mean-aggregation scale into A
        v2f bl = *(const v2f*)(wlp + k);
        c = __builtin_amdgcn_wmma_f32_16x16x4_f32(false, a0, false, bl,
                                                  (short)0, c, false, false);
        v2f a1 = *(const v2f*)(xl + k);
        v2f br = *(const v2f*)(wrp + k);
        c = __builtin_amdgcn_wmma_f32_16x16x4_f32(false, a1, false, br,
                                                  (short)0, c, false, false);
    }

    // epilogue: this lane owns column ncol, rows m0 + r + 8*(lane>>4)
    const int ncol = n0 + lrow;
    const float bb    = b1[ncol];
    const float mu    = mean[ncol];
    const float scale = gamma[ncol] * rsqrtf(var[ncol] + BN_EPS);
    const float bt    = beta[ncol];
    const int rbase = m0 + ((lane >> 4) << 3);
#pragma unroll
    for (int r = 0; r < 8; ++r) {
        float v = c[r] + bb;
        v = fmaxf(v, 0.0f);
        v = (v - mu) * scale + bt;
        h[(rbase + r) * N_HID + ncol] = v;
    }
}

// ---------------- 5) pad W2/b2 from 40 -> 48 rows ----------------
__global__ void sage_padw2_kernel(const float* __restrict__ W2l, const float* __restrict__ W2r,
                                  const float* __restrict__ b2,
                                  float* __restrict__ W2l_p, float* __restrict__ W2r_p,
                                  float* __restrict__ b2_p) {
    int i = blockIdx.x * blockDim.x + threadIdx.x;
    if (i < N_CLS_P * N_HID) {
        int row = i / N_HID;
        W2l_p[i] = (row < N_CLS) ? W2l[i] : 0.0f;   // rows <40 map 1:1
        W2r_p[i] = (row < N_CLS) ? W2r[i] : 0.0f;
    }
    if (i < N_CLS_P) b2_p[i] = (i < N_CLS) ? b2[i] : 0.0f;
}

// ---------------- 6) fused GEMM2: out = agg2*invdeg @ W2l^T + b2 + h @ W2r^T ----------------
// 3 waves/block = the 3 padded col tiles (N_CLS_P = 48); store only first 40 columns
__global__ __launch_bounds__(96) void sage_gemm2_kernel(
    const float* __restrict__ agg2, const float* __restrict__ h,
    const float* __restrict__ W2l_p, const float* __restrict__ W2r_p,
    const float* __restrict__ b2_p,  const float* __restrict__ invdeg,
    float* __restrict__ out) {
    extern __shared__ float smem[];           // [0,2048): agg2 tile, [2048,4096): h tile
    const int lane = threadIdx.x & 31;
    const int wave = threadIdx.x >> 5;        // 0..2
    const int m0 = blockIdx.x * 16;
    const int n0 = wave * 16;
    const int lrow  = lane & 15;
    const int khalf = (lane >> 4) << 1;

    if (wave == 0) {
        tdm_load_tile_16x128(agg2 + m0 * N_HID, 0u);
        __builtin_amdgcn_s_wait_tensorcnt(0);
    } else if (wave == 1) {
        tdm_load_tile_16x128(h + m0 * N_HID, 16u * N_HID * 4u);
        __builtin_amdgcn_s_wait_tensorcnt(0);
    }
    __syncthreads();

    const int brow = n0 + lrow;
    const float idg = invdeg[m0 + lrow];

    const float* aggl = smem + lrow * N_HID + khalf;
    const float* hl   = smem + 16 * N_HID + lrow * N_HID + khalf;
    const float* wlp  = W2l_p + brow * N_HID + khalf;
    const float* wrp  = W2r_p + brow * N_HID + khalf;

    v8f c = {};
#pragma unroll
    for (int k = 0; k < N_HID; k += 4) {
        v2f a0 = *(const v2f*)(aggl + k);
        a0 *= idg;
        v2f bl = *(const v2f*)(wlp + k);
        c = __builtin_amdgcn_wmma_f32_16x16x4_f32(false, a0, false, bl,
                                                  (short)0, c, false, false);
        v2f a1 = *(const v2f*)(hl + k);
        v2f br = *(const v2f*)(wrp + k);
        c = __builtin_amdgcn_wmma_f32_16x16x4_f32(false, a1, false, br,
                                                  (short)0, c, false, false);
    }

    const int ncol = n0 + lrow;
    const float bb = b2_p[ncol];
    const int rbase = m0 + ((lane >> 4) << 3);
    if (ncol < N_CLS) {
#pragma unroll
        for (int r = 0; r < 8; ++r)
            out[(rbase + r) * N_CLS + ncol] = c[r] + bb;
    }
}

// ---------------- launcher ----------------
extern "C" void kernel_launch(void* const* d_in, const int* in_sizes, int n_in,
                              void* d_out, int out_size, void* d_ws, size_t ws_size,
                              hipStream_t stream) {
    const float* x      = (const float*)d_in[0];
    const int*   edge   = (const int*)d_in[1];      // [2, N_EDGES] flat (int32 under JAX)
    const float* W1l    = (const float*)d_in[2];
    const float* b1     = (const float*)d_in[3];
    const float* W1r    = (const float*)d_in[4];
    const float* gamma  = (const float*)d_in[5];
    const float* beta   = (const float*)d_in[6];
    const float* rmean  = (const float*)d_in[7];
    const float* rvar   = (const float*)d_in[8];
    const float* W2l    = (const float*)d_in[9];
    const float* b2     = (const float*)d_in[10];
    const float* W2r    = (const float*)d_in[11];
    float* out = (float*)d_out;

    const int* src = edge;
    const int* dst = edge + N_EDGES;

    float* ws     = (float*)d_ws;
    float* invdeg = ws + WS_INVDEG;
    float* agg1   = ws + WS_AGG1;
    float* agg2   = ws + WS_AGG2;
    float* h      = ws + WS_H;
    float* W2l_p  = ws + WS_W2L;
    float* W2r_p  = ws + WS_W2R;
    float* b2_p   = ws + WS_B2;

    const size_t smem_bytes = 2u * 16u * 128u * sizeof(float);  // 16 KB A-tile staging

    // 1) zero invdeg+agg1+agg2 (contiguous, float4-exact)
    {
        int n4 = WS_ZERO_F / 4;
        sage_zero_kernel<<<(n4 + 255) / 256, 256, 0, stream>>>(ws, n4);
    }
    // 2) degree + invdeg
    sage_deg_kernel<<<(N_EDGES + 255) / 256, 256, 0, stream>>>(dst, invdeg);
    sage_invdeg_kernel<<<(N_NODES + 255) / 256, 256, 0, stream>>>(invdeg);
    // 3) scatter x -> agg1
    sage_scatter_kernel<<<(N_EDGES * 32) / 256, 256, 0, stream>>>(src, dst, x, agg1);
    // 4) GEMM1 + bias + ReLU + BN -> h   (3125 row tiles x 8 col tiles)
    sage_gemm1_kernel<<<N_NODES / 16, 256, smem_bytes, stream>>>(
        agg1, x, W1l, W1r, b1, gamma, beta, rmean, rvar, invdeg, h);
    // 5) scatter h -> agg2 ; pad W2/b2 (independent, overlaps fine on stream)
    sage_padw2_kernel<<<(N_CLS_P * N_HID + 255) / 256, 256, 0, stream>>>(
        W2l, W2r, b2, W2l_p, W2r_p, b2_p);
    sage_scatter_kernel<<<(N_EDGES * 32) / 256, 256, 0, stream>>>(src, dst, h, agg2);
    // 6) GEMM2 + bias -> out (masked to 40 cols)
    sage_gemm2_kernel<<<N_NODES / 16, 96, smem_bytes, stream>>>(
        agg2, h, W2l_p, W2r_p, b2_p, invdeg, out);
}